// GraphInteractionLayer_2061584302794
// MI455X (gfx1250) — compile-verified
//
#include <hip/hip_runtime.h>

typedef float v2f __attribute__((ext_vector_type(2)));
typedef float v8f __attribute__((ext_vector_type(8)));

// Problem constants (from reference setup)
#define B_   8
#define N_   256
#define D_CODE   128
#define D_ATTR   32
#define D_INFO   160   // D_CODE + D_ATTR
#define D_ECODE  128
#define D_EATTR  16
#define ROWS_    (B_ * N_)        // 2048
#define LDX2     288              // node_info(160) | edge_agg(128)

// ---------------------------------------------------------------------------
// Kernel A: pack node_info = [node_code | node_attr] into X2 cols 0..159
// ---------------------------------------------------------------------------
__global__ void pack_node_info_kernel(const float* __restrict__ node_code,
                                      const float* __restrict__ node_attr,
                                      float* __restrict__ X2) {
    const int row = blockIdx.x;          // 0..2047
    const int t   = threadIdx.x;         // 0..159
    float v = (t < D_CODE) ? node_code[(size_t)row * D_CODE + t]
                           : node_attr[(size_t)row * D_ATTR + (t - D_CODE)];
    X2[(size_t)row * LDX2 + t] = v;
}

// ---------------------------------------------------------------------------
// Generic fp32 WMMA GEMM:  Y(2048 x 128) = X(2048 x K, stride ldx) @ W(K x 128)
//                          (+ bias broadcast over rows, if non-null)
// Block = 256 threads = 8 waves; block handles a 16-row M tile, wave w handles
// columns [16w, 16w+16). K must be a multiple of 4.
// A frag (16x4 f32, 2 VGPRs): lane l -> row m0+(l&15), K = 4k0 + 2*(l>>4) + v
// B frag (4x16 f32, 2 VGPRs): lane l -> col c0+(l&15), K = 4k0 + 2*(l>>4) + v
// D (16x16 f32, 8 VGPRs):     lane l -> col c0+(l&15), row m0 + v + 8*(l>>4)
// ---------------------------------------------------------------------------
__global__ void gemm_wmma_f32_kernel(const float* __restrict__ X, int ldx,
                                     const float* __restrict__ W, int K,
                                     const float* __restrict__ bias,
                                     float* __restrict__ Y, int ldy) {
    const int m0   = blockIdx.x * 16;
    const int lane = threadIdx.x & 31;
    const int wave = threadIdx.x >> 5;    // 0..7
    const int n    = lane & 15;
    const int h    = lane >> 4;           // lane half
    const int c    = wave * 16 + n;

    const float* xrow = X + (size_t)(m0 + n) * ldx + 2 * h;

    v8f d = {};
    for (int k = 0; k < K; k += 4) {
        v2f a = *(const v2f*)(xrow + k);
        v2f bm;
        bm[0] = W[(size_t)(k + 2 * h + 0) * D_ECODE + c];
        bm[1] = W[(size_t)(k + 2 * h + 1) * D_ECODE + c];
        d = __builtin_amdgcn_wmma_f32_16x16x4_f32(
                /*neg_a=*/false, a, /*neg_b=*/false, bm,
                /*c_mod=*/(short)0, d, /*reuse_a=*/false, /*reuse_b=*/false);
    }

    const float bv = bias ? bias[c] : 0.0f;
#pragma unroll
    for (int v = 0; v < 8; ++v) {
        const int m = m0 + v + 8 * h;
        Y[(size_t)m * ldy + c] = d[v] + bv;
    }
}

// ---------------------------------------------------------------------------
// Fused edge kernel.  One block per (b,i) pair (2048 blocks, 256 threads).
// For each j-tile (16 senders) and c-tile (16 edge-code channels per wave):
//   D = edge_attr[b,i,j0:j0+16, 0:16] @ W_e1     (4 chained f32 WMMAs, K=16)
//   e = D + Rb[b,i,c] + S[b,j,c]                 (Rb has b_edge folded in)
//   lr = max(e, 0.01*e)                          (leaky relu)
//   acc += lr * adj[b,i,j]                       (aggregate over j)
// Result written into X2 cols 160..287 (edge_agg) for the final node GEMM.
// ---------------------------------------------------------------------------
__global__ void edge_fused_kernel(const float* __restrict__ edge_attr,
                                  const float* __restrict__ adj,
                                  const float* __restrict__ W_edge,  // rows 0..15 used
                                  const float* __restrict__ Rb,      // 2048x128, bias folded
                                  const float* __restrict__ S,       // 2048x128
                                  float* __restrict__ X2) {
    const int bi   = blockIdx.x;          // b*256 + i
    const int b    = bi >> 8;
    const int lane = threadIdx.x & 31;
    const int wave = threadIdx.x >> 5;    // 0..7 -> c tile
    const int n    = lane & 15;
    const int h    = lane >> 4;
    const int c    = wave * 16 + n;

    // Preload W_e1 B-fragments (K=16 -> 4 chunks of 4)
    v2f bfrag[4];
#pragma unroll
    for (int k0 = 0; k0 < 4; ++k0) {
        const int kb = 4 * k0 + 2 * h;
        bfrag[k0][0] = W_edge[(size_t)(kb + 0) * D_ECODE + c];
        bfrag[k0][1] = W_edge[(size_t)(kb + 1) * D_ECODE + c];
    }

    const float rs = Rb[(size_t)bi * D_ECODE + c];   // R + b_edge, constant over j
    const float* ea_base = edge_attr + (size_t)bi * N_ * D_EATTR;
    const float* adj_row = adj + (size_t)bi * N_;
    const float* S_b     = S + (size_t)(b << 8) * D_ECODE;

    float acc[8];
#pragma unroll
    for (int v = 0; v < 8; ++v) acc[v] = 0.0f;

    for (int jt = 0; jt < 16; ++jt) {
        const int j0 = jt * 16;
        // A fragments come straight from global (row j0+n, 16 contiguous floats)
        const float* arow = ea_base + (size_t)(j0 + n) * D_EATTR + 2 * h;

        v8f d = {};
#pragma unroll
        for (int k0 = 0; k0 < 4; ++k0) {
            v2f a = *(const v2f*)(arow + 4 * k0);
            d = __builtin_amdgcn_wmma_f32_16x16x4_f32(
                    false, a, false, bfrag[k0], (short)0, d, false, false);
        }

        // adj fragment: lane l holds adj[j0 + (l&15)]; row of D vgpr v is j0+v+8h
        const float adj_l = adj_row[j0 + n];
#pragma unroll
        for (int v = 0; v < 8; ++v) {
            const int j = j0 + v + 8 * h;
            float e  = d[v] + rs + S_b[(size_t)j * D_ECODE + c];
            float lr = fmaxf(e, 0.01f * e);             // leaky_relu, slope .01
            float av = __shfl(adj_l, v + 8 * h, 32);    // adj[b,i,j]
            acc[v] = fmaf(lr, av, acc[v]);
        }
    }

    // Reduce over the 8 rows held per lane, then fold the two lane halves
    float s = 0.0f;
#pragma unroll
    for (int v = 0; v < 8; ++v) s += acc[v];
    s += __shfl_xor(s, 16, 32);

    if (h == 0) {
        X2[(size_t)bi * LDX2 + D_INFO + c] = s;   // edge_agg into cols 160..287
    }
}

// ---------------------------------------------------------------------------
// Launcher
// ---------------------------------------------------------------------------
extern "C" void kernel_launch(void* const* d_in, const int* in_sizes, int n_in,
                              void* d_out, int out_size, void* d_ws, size_t ws_size,
                              hipStream_t stream) {
    const float* node_code = (const float*)d_in[0];
    const float* node_attr = (const float*)d_in[1];
    const float* edge_attr = (const float*)d_in[2];
    const float* adj       = (const float*)d_in[3];
    const float* W_edge    = (const float*)d_in[4];  // 336 x 128
    const float* b_edge    = (const float*)d_in[5];
    const float* W_node    = (const float*)d_in[6];  // 288 x 128
    const float* b_node    = (const float*)d_in[7];
    float* out = (float*)d_out;

    char* ws = (char*)d_ws;
    float* X2 = (float*)ws;                                        // 2048 x 288
    float* Rb = (float*)(ws + (size_t)ROWS_ * LDX2 * sizeof(float));     // 2048 x 128
    float* S  = Rb + (size_t)ROWS_ * D_ECODE;                      // 2048 x 128

    // 1) pack node_info into X2[:, 0:160]
    pack_node_info_kernel<<<ROWS_, D_INFO, 0, stream>>>(node_code, node_attr, X2);

    // 2) R(+bias) = node_info @ W_edge[16:176] + b_edge ; S = node_info @ W_edge[176:336]
    gemm_wmma_f32_kernel<<<ROWS_ / 16, 256, 0, stream>>>(
        X2, LDX2, W_edge + 16 * D_ECODE, D_INFO, b_edge, Rb, D_ECODE);
    gemm_wmma_f32_kernel<<<ROWS_ / 16, 256, 0, stream>>>(
        X2, LDX2, W_edge + 176 * D_ECODE, D_INFO, nullptr, S, D_ECODE);

    // 3) fused edge MLP + leaky_relu + adj-weighted aggregation -> X2[:, 160:288]
    edge_fused_kernel<<<ROWS_, 256, 0, stream>>>(edge_attr, adj, W_edge, Rb, S, X2);

    // 4) new_node_code = [node_info | edge_agg] @ W_node + b_node
    gemm_wmma_f32_kernel<<<ROWS_ / 16, 256, 0, stream>>>(
        X2, LDX2, W_node, LDX2, b_node, out, D_CODE);
}